// FrameGraph_5059471474795
// MI455X (gfx1250) — compile-verified
//
#include <hip/hip_runtime.h>
#include <hip/hip_bf16.h>
#include <math.h>

// ---------------- problem constants (from reference) ----------------
#define NN_  4096
#define FF_  512
#define KK_  8
#define NK_  (NN_ * KK_)

static constexpr float IMG_W   = 1920.0f;
static constexpr float IMG_H   = 1080.0f;
static constexpr float THRESH2 = 108.0f * 108.0f;   // (0.1*min(W,H))^2
static constexpr float BIGD2   = 1.0e30f;           // sentinel in d^2 space
static constexpr float VALID_CUT = 1.0e29f;
static constexpr float EPSF    = 1.0e-12f;
static constexpr float RAD2DEG = 57.29577951308232f;

// output layout (floats), concatenated in reference return order
static constexpr size_t OFF_NODE  = 0;                         // 4096*532
static constexpr size_t OFF_EIDX  = (size_t)NN_ * 532;         // 2*32768
static constexpr size_t OFF_EATTR = OFF_EIDX + 2 * (size_t)NK_;// 32768*1029
static constexpr size_t OFF_OCC   = OFF_EATTR + (size_t)NK_ * 1029;
static constexpr size_t OFF_MASK  = OFF_OCC + (size_t)NN_;

typedef float v2f_t __attribute__((ext_vector_type(2)));
typedef float v8f_t __attribute__((ext_vector_type(8)));

// ---------------- kernel 0: per-box prep ----------------
// cxy[i] = (cx,cy);  bvec[i] = (-2cx, -2cy, 1, r) with r = cx^2+cy^2
__global__ __launch_bounds__(256) void fg_prep_kernel(
    const float4* __restrict__ pos4, float2* __restrict__ cxy,
    float4* __restrict__ bvec) {
  int i = blockIdx.x * blockDim.x + threadIdx.x;
  if (i >= NN_) return;
  float4 p = pos4[i];
  float cx = 0.5f * (p.x + p.z);
  float cy = 0.5f * (p.y + p.w);
  float r  = cx * cx + cy * cy;
  cxy[i]  = make_float2(cx, cy);
  bvec[i] = make_float4(-2.0f * cx, -2.0f * cy, 1.0f, r);
}

// ---------------- kernel 1: reid row L2-normalize ----------------
// one wave per row; writes normalized reid to ws and node_feat[:,0:512]
__global__ __launch_bounds__(256) void fg_norm_kernel(
    const float* __restrict__ reid, float* __restrict__ reidn,
    float* __restrict__ d_out) {
  int row  = blockIdx.x * 8 + (threadIdx.x >> 5);
  int lane = threadIdx.x & 31;
  const float4* r4 = (const float4*)(reid + (size_t)row * FF_);
  float4 v[4];
  float s = 0.0f;
#pragma unroll
  for (int w = 0; w < 4; ++w) {
    v[w] = r4[lane + w * 32];
    s += v[w].x * v[w].x + v[w].y * v[w].y + v[w].z * v[w].z + v[w].w * v[w].w;
  }
#pragma unroll
  for (int m = 16; m >= 1; m >>= 1) s += __shfl_xor(s, m, 32);
  float inv = 1.0f / (sqrtf(s) + EPSF);
  float4* wn = (float4*)(reidn + (size_t)row * FF_);
  float4* on = (float4*)(d_out + OFF_NODE + (size_t)row * 532);  // 532 % 4 == 0
#pragma unroll
  for (int w = 0; w < 4; ++w) {
    float4 o = make_float4(v[w].x * inv, v[w].y * inv, v[w].z * inv, v[w].w * inv);
    wn[lane + w * 32] = o;
    on[lane + w * 32] = o;
  }
}

// ---------------- kernel 2: WMMA distance tiles + per-row top-8 ----------------
// block = 128 threads = 4 waves, handles 16 rows.  Each wave WMMAs column
// tiles t ≡ wave (mod 4):  d2 = A(16x4) x B(4x16) with
//   A[m] = [cx, cy, r, 1]   B[:,j] = [-2cx, -2cy, 1, r]
// Software-pipelined B-fragment loads; b128 candidate fetch from LDS.
__device__ __forceinline__ v2f_t fg_load_bfrag(const float4* __restrict__ bvec,
                                               int col0, int lane) {
  // lanes 0-15 take (B[0][j],B[1][j]) = (-2cx,-2cy); lanes 16-31 take (1, r)
  const float* bp =
      (const float*)(bvec + col0 + (lane & 15)) + ((lane >> 4) << 1);
  v2f_t b;
  b.x = bp[0];
  b.y = bp[1];
  return b;
}

__global__ __launch_bounds__(128) void fg_dist_topk_kernel(
    const float4* __restrict__ pos4, const float2* __restrict__ cxy,
    const float4* __restrict__ bvec, float* __restrict__ d_out,
    int* __restrict__ nbidx, float* __restrict__ nbd2) {
  __shared__ float tile[4][16][20];   // 80B row stride: 16B-aligned half-rows
  __shared__ float candv[16][64];
  __shared__ int   candi[16][64];

  const int tid  = threadIdx.x;
  const int wave = tid >> 5;
  const int lane = tid & 31;
  const int row0 = blockIdx.x * 16;

  // A fragment: 32-bit 16x4 layout -> lanes 0-15: (K0,K1), lanes 16-31: (K2,K3)
  v2f_t a;
  if (lane < 16) {
    float2 cc = cxy[row0 + lane];
    a.x = cc.x; a.y = cc.y;
  } else {
    float4 bb = bvec[row0 + lane - 16];
    a.x = bb.w;   // r
    a.y = 1.0f;
  }

  const int sel_row = lane >> 1;
  const int sel_sub = lane & 1;
  const int lr = (lane >> 4) * 8;
  const int lc = lane & 15;
  float bd[8]; int bi[8];
#pragma unroll
  for (int q = 0; q < 8; ++q) { bd[q] = 3.0e30f; bi[q] = 0; }

  v2f_t bcur = fg_load_bfrag(bvec, wave * 16, lane);

  for (int t = wave; t < NN_ / 16; t += 4) {
    const int col0 = t * 16;
    // prefetch next tile's B fragment (uniform branch per wave)
    v2f_t bnext = bcur;
    if (t + 4 < NN_ / 16) bnext = fg_load_bfrag(bvec, (t + 4) * 16, lane);

    v8f_t acc = {};
    acc = __builtin_amdgcn_wmma_f32_16x16x4_f32(false, a, false, bcur, (short)0,
                                                acc, false, false);
    float v[8];
#pragma unroll
    for (int r = 0; r < 8; ++r) {
      float x = fmaxf(acc[r], EPSF);
      v[r] = (x > THRESH2) ? BIGD2 : x;
    }
    // diagonal only exists in the aligned tile (wave-uniform condition)
    if (col0 == row0) {
      const int dr = lc - lr;
#pragma unroll
      for (int r = 0; r < 8; ++r)
        if (dr == r) v[r] = BIGD2;
    }
#pragma unroll
    for (int r = 0; r < 8; ++r) tile[wave][lr + r][lc] = v[r];

    asm volatile("s_wait_dscnt 0" ::: "memory");  // per-wave RAW LDS ordering

    // 2 lanes per row fetch 8 candidates as 2x b128, insert from registers
    {
      const float4* trow = (const float4*)&tile[wave][sel_row][sel_sub * 8];
      float4 c0 = trow[0];
      float4 c1 = trow[1];
      float cv[8] = {c0.x, c0.y, c0.z, c0.w, c1.x, c1.y, c1.z, c1.w};
#pragma unroll
      for (int q = 0; q < 8; ++q) {
        const float cvq = cv[q];
        if (cvq < bd[7]) {
          bd[7] = cvq; bi[7] = col0 + sel_sub * 8 + q;
#pragma unroll
          for (int p = 7; p >= 1; --p) {
            if (bd[p] < bd[p - 1]) {
              float tv = bd[p]; bd[p] = bd[p - 1]; bd[p - 1] = tv;
              int   ti = bi[p]; bi[p] = bi[p - 1]; bi[p - 1] = ti;
            }
          }
        }
      }
    }
    asm volatile("" ::: "memory");
    bcur = bnext;
  }

  // dump 8 partial lists per row (4 waves x 2 sublanes)
  const int slot = wave * 16 + sel_sub * 8;
#pragma unroll
  for (int q = 0; q < 8; ++q) {
    candv[sel_row][slot + q] = bd[q];
    candi[sel_row][slot + q] = bi[q];
  }
  __syncthreads();

  if (tid < 16) {
    const int i = row0 + tid;
    float fd[8]; int fidx[8];
#pragma unroll
    for (int q = 0; q < 8; ++q) { fd[q] = candv[tid][q]; fidx[q] = candi[tid][q]; }
    for (int s = 8; s < 64; ++s) {
      const float v = candv[tid][s];
      if (v < fd[7]) {
        fd[7] = v; fidx[7] = candi[tid][s];
#pragma unroll
        for (int p = 7; p >= 1; --p) {
          if (fd[p] < fd[p - 1]) {
            float tv = fd[p]; fd[p] = fd[p - 1]; fd[p - 1] = tv;
            int   ti = fidx[p]; fidx[p] = fidx[p - 1]; fidx[p - 1] = ti;
          }
        }
      }
    }

    // ---- per-row outputs ----
    float4 p = pos4[i];
    float w  = p.z - p.x, h = p.w - p.y;
    float cx = 0.5f * (p.x + p.z), cy = 0.5f * (p.y + p.w);
    float* nf = d_out + OFF_NODE + (size_t)i * 532;
    nf[512] = cx / IMG_W; nf[513] = cy / IMG_H;
    nf[514] = w  / IMG_W; nf[515] = h  / IMG_H;

    bool valid[8]; float vx[8], vy[8];
#pragma unroll
    for (int q = 0; q < 8; ++q) {
      valid[q] = fd[q] < VALID_CUT;
      float dist = valid[q] ? sqrtf(fmaxf(fd[q], EPSF)) : 0.0f;
      nf[516 + q] = dist / IMG_H;
      float2 cn = cxy[fidx[q]];
      vx[q] = cn.x - cx; vy[q] = cn.y - cy;
    }
#pragma unroll
    for (int q = 0; q < 7; ++q) {
      float dotp = vx[q] * vx[q + 1] + vy[q] * vy[q + 1];
      float n1 = sqrtf(vx[q] * vx[q] + vy[q] * vy[q]);
      float n2 = sqrtf(vx[q + 1] * vx[q + 1] + vy[q + 1] * vy[q + 1]);
      float ca = dotp / (n1 * n2 + EPSF);
      ca = fminf(fmaxf(ca, -1.0f + 1e-6f), 1.0f - 1e-6f);
      float ang = acosf(ca) * RAD2DEG;
      ang = valid[q + 1] ? ang : 0.0f;
      nf[524 + q] = ang * (1.0f / 360.0f);
    }
    nf[531] = 0.0f;

    // occlusion vs nearest neighbor
    {
      float4 pn = pos4[fidx[0]];
      float ix1 = fmaxf(p.x, pn.x), iy1 = fmaxf(p.y, pn.y);
      float ix2 = fminf(p.z, pn.z), iy2 = fminf(p.w, pn.w);
      float ov = fmaxf(ix2 - ix1, 0.0f) * fmaxf(iy2 - iy1, 0.0f);
      d_out[OFF_OCC + i] =
          (valid[0] && (ov > w * h * 0.5f)) ? 1.0f : 0.0f;
    }
#pragma unroll
    for (int q = 0; q < 8; ++q) {
      nbidx[i * KK_ + q] = fidx[q];
      nbd2[i * KK_ + q]  = fd[q];
      d_out[OFF_MASK + (size_t)i * KK_ + q] = valid[q] ? 1.0f : 0.0f;
    }
  }
}

// ---------------- kernel 3: edge_index + edge_attr ----------------
// one wave per edge; 8 edges per 256-thread block
__global__ __launch_bounds__(256) void fg_edge_kernel(
    const float4* __restrict__ pos4, const float2* __restrict__ cxy,
    const float* __restrict__ reidn, const int* __restrict__ nbidx,
    const float* __restrict__ nbd2, float* __restrict__ d_out) {
  const int e    = blockIdx.x * 8 + (threadIdx.x >> 5);
  const int lane = threadIdx.x & 31;
  const int i    = e >> 3;                  // src
  const float d2 = nbd2[e];
  const bool valid = d2 < VALID_CUT;
  const int dst    = valid ? nbidx[e] : 0;
  const float m    = valid ? 1.0f : 0.0f;

  float* ea = d_out + OFF_EATTR + (size_t)e * 1029;

  if (lane == 0) {
    d_out[OFF_EIDX + e]        = (float)i;
    d_out[OFF_EIDX + NK_ + e]  = (float)dst;
    float2 cs = cxy[i], cd = cxy[dst];
    float4 ps = pos4[i], pd = pos4[dst];
    float xd = (cs.x - cd.x) / IMG_W;
    float yd = (cs.y - cd.y) / IMG_H;
    float ws_ = ps.z - ps.x, hs = ps.w - ps.y;
    float wd = pd.z - pd.x, hd = pd.w - pd.y;
    float areas = ws_ * hs, aread = wd * hd;
    float ix1 = fmaxf(ps.x, pd.x), iy1 = fmaxf(ps.y, pd.y);
    float ix2 = fminf(ps.z, pd.z), iy2 = fminf(ps.w, pd.w);
    float inter = fmaxf(ix2 - ix1, 0.0f) * fmaxf(iy2 - iy1, 0.0f);
    float iou = inter / (areas + aread - inter + EPSF);
    ea[0] = xd * m;
    ea[1] = yd * m;
    ea[2] = iou * m;
    ea[3] = logf(ws_ / wd) * m;
    ea[4] = logf(hs / hd) * m;
  }

  const float4* rs4 = (const float4*)(reidn + (size_t)i * FF_);
  const float4* rd4 = (const float4*)(reidn + (size_t)dst * FF_);
#pragma unroll
  for (int w4 = 0; w4 < 4; ++w4) {
    const int t4 = lane + w4 * 32;        // 0..127 float4 chunks
    float4 vs = rs4[t4];
    float4 vd = rd4[t4];
    const int bs = 5 + t4 * 4;
    ea[bs + 0] = vs.x * m; ea[bs + 1] = vs.y * m;
    ea[bs + 2] = vs.z * m; ea[bs + 3] = vs.w * m;
    const int bdp = 5 + FF_ + t4 * 4;
    ea[bdp + 0] = vd.x * m; ea[bdp + 1] = vd.y * m;
    ea[bdp + 2] = vd.z * m; ea[bdp + 3] = vd.w * m;
  }
}

// ---------------- launcher ----------------
extern "C" void kernel_launch(void* const* d_in, const int* in_sizes, int n_in,
                              void* d_out, int out_size, void* d_ws,
                              size_t ws_size, hipStream_t stream) {
  const float* reid = (const float*)d_in[0];   // (N,512)
  const float* pos  = (const float*)d_in[1];   // (N,4) tlbr
  // d_in[2] (confs) is unused by the reference forward
  float* out = (float*)d_out;
  char* ws = (char*)d_ws;

  float2* cxy   = (float2*)(ws);                    //  32 KB
  float4* bvec  = (float4*)(ws + (32u << 10));      //  64 KB
  int*    nbidx = (int*)  (ws + (96u << 10));       // 128 KB
  float*  nbd2  = (float*)(ws + (224u << 10));      // 128 KB
  float*  reidn = (float*)(ws + (352u << 10));      //   8 MB

  const float4* pos4 = (const float4*)pos;

  fg_prep_kernel<<<(NN_ + 255) / 256, 256, 0, stream>>>(pos4, cxy, bvec);
  fg_norm_kernel<<<NN_ / 8, 256, 0, stream>>>(reid, reidn, out);
  fg_dist_topk_kernel<<<NN_ / 16, 128, 0, stream>>>(pos4, cxy, bvec, out,
                                                    nbidx, nbd2);
  fg_edge_kernel<<<NK_ / 8, 256, 0, stream>>>(pos4, cxy, reidn, nbidx, nbd2,
                                              out);
}